// GatedPool2d_63402307224337
// MI455X (gfx1250) — compile-verified
//
#include <hip/hip_runtime.h>
#include <hip/hip_bf16.h>
#include <stdint.h>

typedef float v2f __attribute__((ext_vector_type(2)));
typedef float v8f __attribute__((ext_vector_type(8)));

#define B_  32
#define C_  128
#define H_  112
#define W_  112
#define OH  56
#define OW  56
#define TILE_H 8    // output rows per block (one per wave)
#define TILE_W 16   // output cols per block (WMMA M dimension)
#define TLH 7       // 56 / 8
#define TLW 4       // ceil(56 / 16); last tile is half-valid

// Low 32 bits of a generic pointer into LDS == byte address within the
// workgroup's LDS allocation (AMDGPU generic-pointer encoding).
__device__ __forceinline__ uint32_t lds_byte_addr(const void* p) {
  return (uint32_t)(uintptr_t)p;
}

__global__ __launch_bounds__(256)
void gatedpool2d_kernel(const float* __restrict__ x,
                        const float* __restrict__ wt,
                        float* __restrict__ out)
{
  __shared__ float lds_w[C_ * 4];           // 512 floats = 2 KB
  __shared__ float lds_mask[TILE_H * TILE_W];

  const int t   = threadIdx.x;
  const int blk = blockIdx.x;
  const int b   = blk / (TLH * TLW);
  const int tt  = blk % (TLH * TLW);
  const int th  = tt / TLW;
  const int tw  = tt % TLW;

  const int in_row0 = th * (2 * TILE_H);    // first input row of tile
  const int in_col0 = tw * (2 * TILE_W);    // first input col of tile

  // ---- Phase 1: weights -> LDS via CDNA5 async DMA path --------------------
  if (t < 128) {
    uint64_t ga = (uint64_t)(uintptr_t)(wt + t * 4);
    uint32_t la = lds_byte_addr(&lds_w[t * 4]);
    asm volatile("global_load_async_to_lds_b128 %0, %1, off"
                 :: "v"(la), "v"(ga) : "memory");
  }
  asm volatile("s_wait_asynccnt 0x0" ::: "memory");
  __syncthreads();

  // ---- Phase 2: gate logits via V_WMMA_F32_16X16X4_F32 ---------------------
  // Wave g computes logits for output row g of the tile (16 columns).
  // A (16x4): lane L: row m = L&15 (= output col), K-half h = L>>4 (= kh).
  //   per-lane float2 = x[c][2g+h][2m .. 2m+1]  (coalesced 128B per half-wave)
  // B (4x16): per-lane float2 = (w[c][2h], w[c][2h+1]) replicated over all 16
  //   columns -> every D column equals the logit vector for this tile row.
  const int lane = t & 31;
  const int g    = t >> 5;
  const int m    = lane & 15;
  const int h    = lane >> 4;

  int acol = in_col0 + 2 * m;
  if (acol > W_ - 2) acol = W_ - 2;          // clamp for partial tiles
  const float* aptr = x + ((size_t)b * C_) * (H_ * W_)
                        + (size_t)(in_row0 + 2 * g + h) * W_ + acol;
  const float* bptr = lds_w + 2 * h;

  v8f acc0 = {0.f, 0.f, 0.f, 0.f, 0.f, 0.f, 0.f, 0.f};
  v8f acc1 = acc0;
  for (int c = 0; c < C_; c += 2) {
    v2f a0 = *(const v2f*)(aptr);
    v2f b0 = *(const v2f*)(bptr);
    acc0 = __builtin_amdgcn_wmma_f32_16x16x4_f32(false, a0, false, b0,
                                                 (short)0, acc0, false, false);
    v2f a1 = *(const v2f*)(aptr + H_ * W_);
    v2f b1 = *(const v2f*)(bptr + 4);
    acc1 = __builtin_amdgcn_wmma_f32_16x16x4_f32(false, a1, false, b1,
                                                 (short)0, acc1, false, false);
    aptr += 2 * (H_ * W_);
    bptr += 8;
  }
  v8f acc = acc0 + acc1;

  // D layout: VGPR r, lanes 0-15 -> M=r (cols 0..7); lanes 16-31 -> M=8+r.
  if (lane == 0) {
#pragma unroll
    for (int r = 0; r < 8; ++r)
      lds_mask[g * TILE_W + r] = 1.f / (1.f + __expf(-acc[r]));
  } else if (lane == 16) {
#pragma unroll
    for (int r = 0; r < 8; ++r)
      lds_mask[g * TILE_W + 8 + r] = 1.f / (1.f + __expf(-acc[r]));
  }
  __syncthreads();

  // ---- Phase 3: 2x2 max/avg pool + sigmoid blend ---------------------------
  // Re-reads the block's 256 KB input window; it was just streamed by phase 2
  // and the combined working set of resident blocks fits in the 192 MB L2,
  // so this costs no extra HBM traffic.
  const float* xb = x + ((size_t)b * C_) * (H_ * W_);
#pragma unroll 4
  for (int i = 0; i < 64; ++i) {
    int j    = t + 256 * i;
    int col  = j & 15;
    int row  = (j >> 4) & 7;
    int c    = j >> 7;
    int icol = in_col0 + 2 * col;
    int ocol = tw * TILE_W + col;
    if (icol > W_ - 2) icol = W_ - 2;
    const float* p0 = xb + (size_t)c * (H_ * W_)
                         + (size_t)(in_row0 + 2 * row) * W_ + icol;
    v2f a = *(const v2f*)(p0);
    v2f d = *(const v2f*)(p0 + W_);
    float mx = fmaxf(fmaxf(a.x, a.y), fmaxf(d.x, d.y));
    float av = (a.x + a.y + d.x + d.y) * 0.25f;
    float mk = lds_mask[row * TILE_W + col];
    if (ocol < OW) {
      out[((size_t)(b * C_ + c) * OH + (th * TILE_H + row)) * OW + ocol]
          = mk * mx + (1.f - mk) * av;
    }
  }
}

extern "C" void kernel_launch(void* const* d_in, const int* in_sizes, int n_in,
                              void* d_out, int out_size, void* d_ws, size_t ws_size,
                              hipStream_t stream) {
  const float* x  = (const float*)d_in[0];   // (32,128,112,112) f32
  const float* wt = (const float*)d_in[1];   // (128,2,2) f32
  float* out = (float*)d_out;                // (32,128,56,56) f32
  (void)in_sizes; (void)n_in; (void)out_size; (void)d_ws; (void)ws_size;

  dim3 grid(B_ * TLH * TLW);   // 32 * 7 * 4 = 896 blocks
  dim3 block(256);             // 8 waves: one per output tile row
  gatedpool2d_kernel<<<grid, block, 0, stream>>>(x, wt, out);
}